// CPainted_11665131176542
// MI455X (gfx1250) — compile-verified
//
#include <hip/hip_runtime.h>
#include <stdint.h>

#define HH 480
#define WW 640
#define HWSZ (HH * WW)
#define NB 16
#define KTOP 400
#define HC 60
#define WC 80
#define CDESC 256
#define SORTN 8192  // power of two, >= worst-case candidates (~HW/49 ~ 6.3K)

typedef uint32_t v4u __attribute__((ext_vector_type(4)));
typedef uint32_t v8u __attribute__((ext_vector_type(8)));

// ---------------- CDNA5 async global->LDS + TDM helpers ----------------
// GLOBAL_LOAD_ASYNC_TO_LDS_B32/B64: copies global data at vaddr into LDS at the VDST offset; tracked by ASYNCcnt.
__device__ __forceinline__ void async_g2l_b32(const float* g, uint32_t lds_off) {
  asm volatile("global_load_async_to_lds_b32 %0, %1, off"
               :: "v"(lds_off), "v"(g)
               : "memory");
}
__device__ __forceinline__ void async_g2l_b64(const void* g, uint32_t lds_off) {
  asm volatile("global_load_async_to_lds_b64 %0, %1, off"
               :: "v"(lds_off), "v"(g)
               : "memory");
}
__device__ __forceinline__ void wait_async0() {
  asm volatile("s_wait_asynccnt 0" ::: "memory");
}
// TENSOR_LOAD_TO_LDS, 2-SGPR-group form (tensors up to 2D); tracked by TENSORcnt.
__device__ __forceinline__ void tdm_load_2d(v4u g0, v8u g1) {
  asm volatile("tensor_load_to_lds %0, %1"
               :: "s"(g0), "s"(g1)
               : "memory");
}

// ---------------- Kernel 1: 7x7 NMS + border mask + candidate emit ----------------
// grid: (20*15 tiles, 16 batches), block: 256. Tile = 32x32 output, 38x38 halo.
// Interior tiles: one TDM descriptor DMAs the whole 38x38 halo (fully in-bounds).
// Edge tiles: per-lane clamped async b32 loads (clamped values only feed border-masked px).
__global__ __launch_bounds__(256) void nms_kernel(const float* __restrict__ heat,
                                                  unsigned* __restrict__ counters,
                                                  unsigned long long* __restrict__ cand,
                                                  int cap) {
  __shared__ float tin[38 * 38];   // halo tile, packed stride 38 (TDM packs rows)
  __shared__ float tmid[38 * 32];  // horizontal 7-max

  const int tid = threadIdx.x;
  const int b = blockIdx.y;
  const int tileX = blockIdx.x % 20;
  const int tileY = blockIdx.x / 20;
  const int bx = tileX * 32, by = tileY * 32;
  const float* hb = heat + (size_t)b * HWSZ;

  const bool interior = (tileX > 0) && (tileX < 19) && (tileY > 0) && (tileY < 14);
  if (interior) {
    // Halo [by-3, by+35) x [bx-3, bx+35) is fully inside the 480x640 image.
    if (tid < 32) {  // one wave issues the DMA (TDM ignores EXEC; one issue per wave)
      const uint64_t ga = (uint64_t)(uintptr_t)(hb + (size_t)(by - 3) * WW + (bx - 3));
      const uint32_t lds = (uint32_t)(uintptr_t)(&tin[0]);
      v4u g0;
      g0.x = 1u;                                            // count=1, user mode
      g0.y = lds;                                           // lds_addr
      g0.z = (uint32_t)ga;                                  // global_addr[31:0]
      g0.w = ((uint32_t)(ga >> 32) & 0x01FFFFFFu) | (2u << 30);  // addr[56:32] | type=2
      v8u g1;
      g1.s0 = (2u << 16);                                   // data_size=4B; no mask/pad
      g1.s1 = (uint32_t)(WW & 0xFFFF) << 16;                // tensor_dim0[15:0]
      g1.s2 = (uint32_t)(WW >> 16) | ((uint32_t)(HH & 0xFFFF) << 16);  // dim0 hi, dim1 lo
      g1.s3 = (uint32_t)(HH >> 16) | (38u << 16);           // dim1 hi, tile_dim0=38
      g1.s4 = 38u;                                          // tile_dim1=38, tile_dim2=0
      g1.s5 = (uint32_t)WW;                                 // tensor_dim0_stride=640
      g1.s6 = 0u;
      g1.s7 = 0u;
      tdm_load_2d(g0, g1);
      __builtin_amdgcn_s_wait_tensorcnt(0);
    }
  } else {
    for (int l = tid; l < 38 * 38; l += 256) {
      int ly = l / 38, lx = l - ly * 38;
      int gy = by - 3 + ly; gy = gy < 0 ? 0 : (gy > HH - 1 ? HH - 1 : gy);
      int gx = bx - 3 + lx; gx = gx < 0 ? 0 : (gx > WW - 1 ? WW - 1 : gx);
      uint32_t off = (uint32_t)(uintptr_t)(&tin[ly * 38 + lx]);
      async_g2l_b32(hb + (size_t)gy * WW + gx, off);
    }
    wait_async0();
  }
  __syncthreads();

  // Horizontal 7-max: 38 rows x 32 cols
  for (int e = tid; e < 38 * 32; e += 256) {
    int r = e >> 5, c = e & 31;
    const float* row = &tin[r * 38 + c];
    float m = row[0];
#pragma unroll
    for (int j = 1; j < 7; ++j) m = fmaxf(m, row[j]);
    tmid[r * 32 + c] = m;
  }
  __syncthreads();

  // Vertical 7-max + detect; each thread covers 4 pixels.
  const int x = tid & 31;
  const int ty = tid >> 5;
#pragma unroll
  for (int i = 0; i < 4; ++i) {
    int y = ty + 8 * i;
    float p = tmid[y * 32 + x];
#pragma unroll
    for (int j = 1; j < 7; ++j) p = fmaxf(p, tmid[(y + j) * 32 + x]);
    float h = tin[(y + 3) * 38 + (x + 3)];
    int gy = by + y, gx = bx + x;
    bool ok = (gy >= 4) && (gy < HH - 4) && (gx >= 4) && (gx < WW - 4) &&
              (h >= p) && (h > 0.0f);
    if (ok) {
      unsigned pos = atomicAdd(&counters[b], 1u);
      if (pos < (unsigned)cap) {
        unsigned idx = (unsigned)(gy * WW + gx);
        unsigned sb = __float_as_uint(h);  // h > 0 => bit pattern is order-preserving
        // key: (score desc, idx asc) under descending u64 sort
        cand[(size_t)b * cap + pos] =
            ((unsigned long long)sb << 32) | (unsigned long long)(~idx);
      }
    }
  }
}

// ---------------- Kernel 2: per-batch descending bitonic sort (in LDS) + top-400 emit ---------
// grid: 16 blocks, block: 256. 64KB LDS key buffer (CDNA5 WGP has 320KB).
__global__ __launch_bounds__(256) void topk_sort_kernel(const float* __restrict__ heat,
                                                        const unsigned* __restrict__ counters,
                                                        const unsigned long long* __restrict__ cand,
                                                        int cap,
                                                        float* __restrict__ out) {
  __shared__ unsigned long long sbuf[SORTN];  // 64 KB
  const int b = blockIdx.x;
  const int tid = threadIdx.x;
  const unsigned long long* buf = cand + (size_t)b * cap;
  unsigned cnt = counters[b];
  if (cnt > (unsigned)cap) cnt = (unsigned)cap;

  // Async-DMA the candidate keys into LDS; zero-pad the rest while the DMA is in flight.
  for (int i = tid; i < SORTN; i += 256) {
    if (i < (int)cnt) {
      uint32_t off = (uint32_t)(uintptr_t)(&sbuf[i]);
      async_g2l_b64(&buf[i], off);
    } else {
      sbuf[i] = 0ull;  // pads sort below any real (score>0) key
    }
  }
  wait_async0();
  __syncthreads();

  // Descending bitonic sort over SORTN keys, entirely in LDS.
  for (int k = 2; k <= SORTN; k <<= 1) {
    for (int j = k >> 1; j > 0; j >>= 1) {
      for (int e = tid; e < SORTN; e += 256) {
        int p = e ^ j;
        if (p > e) {
          unsigned long long a = sbuf[e], c = sbuf[p];
          bool up = ((e & k) == 0);
          if (up ? (a < c) : (a > c)) { sbuf[e] = c; sbuf[p] = a; }
        }
      }
      __syncthreads();
    }
  }

  // Emit kpts + scores (jax.top_k semantics; pad path mimics tie-by-lowest-index zeros).
  const float* hb = heat + (size_t)b * HWSZ;
  for (int i = tid; i < KTOP; i += 256) {
    unsigned idx;
    if (i < (int)cnt) {
      unsigned long long key = sbuf[i];
      idx = ~((unsigned)key);
    } else {
      idx = (unsigned)(i - (int)cnt);
    }
    float sc = hb[idx];  // raw heatmap, as reference
    int y = idx / WW, xx = idx - y * WW;
    float* kp = out + ((size_t)b * KTOP + i) * 2;
    kp[0] = (float)xx + 0.5f;
    kp[1] = (float)y + 0.5f;
    out[(size_t)NB * KTOP * 2 + (size_t)b * KTOP + i] = sc;
  }
}

// ---------------- Kernel 3: bilinear descriptor sampling + L2 normalize ----------------
// grid: 16*400 blocks (one keypoint each), block: 256 (one channel each).
// Descriptor array (78.6MB) fits in the 192MB L2, so the strided channel gathers hit L2.
__global__ __launch_bounds__(256) void sample_kernel(const float* __restrict__ desc,
                                                     const float* __restrict__ kpts,
                                                     float* __restrict__ sampled) {
  __shared__ float red[8];
  const int bk = blockIdx.x;
  const int b = bk / KTOP;
  const int k = bk - b * KTOP;
  const int c = threadIdx.x;

  const float kx = kpts[((size_t)b * KTOP + k) * 2 + 0];
  const float ky = kpts[((size_t)b * KTOP + k) * 2 + 1];
  // reference: ((kpt - S/2 + 0.5)/scale)*2 - 1 ; then (v+1)*0.5*(dim-1)
  float nx = (kx - 4.0f + 0.5f) / 635.5f; nx = nx * 2.0f - 1.0f;
  float ny = (ky - 4.0f + 0.5f) / 475.5f; ny = ny * 2.0f - 1.0f;
  const float gx = (nx + 1.0f) * 0.5f * (float)(WC - 1);
  const float gy = (ny + 1.0f) * 0.5f * (float)(HC - 1);
  const float x0f = floorf(gx), y0f = floorf(gy);
  const float wx = gx - x0f, wy = gy - y0f;
  int x0 = (int)x0f; x0 = x0 < 0 ? 0 : (x0 > WC - 1 ? WC - 1 : x0);
  int x1 = x0 + 1 > WC - 1 ? WC - 1 : x0 + 1;
  int y0 = (int)y0f; y0 = y0 < 0 ? 0 : (y0 > HC - 1 ? HC - 1 : y0);
  int y1 = y0 + 1 > HC - 1 ? HC - 1 : y0 + 1;

  const float* db = desc + ((size_t)b * CDESC + c) * (HC * WC);
  const float d00 = db[y0 * WC + x0];
  const float d01 = db[y0 * WC + x1];
  const float d10 = db[y1 * WC + x0];
  const float d11 = db[y1 * WC + x1];
  const float o = d00 * (1.0f - wx) * (1.0f - wy) + d01 * wx * (1.0f - wy) +
                  d10 * (1.0f - wx) * wy + d11 * wx * wy;

  // wave32 butterfly + single LDS combine for sum over 256 channels
  float v = o * o;
#pragma unroll
  for (int s = 16; s > 0; s >>= 1) v += __shfl_xor(v, s, 32);
  if ((c & 31) == 0) red[c >> 5] = v;
  __syncthreads();
  float tot = 0.0f;
#pragma unroll
  for (int w = 0; w < 8; ++w) tot += red[w];
  const float inv = 1.0f / sqrtf(tot + 1e-12f);
  sampled[((size_t)b * CDESC + c) * KTOP + k] = o * inv;
}

// ---------------- host launch ----------------
extern "C" void kernel_launch(void* const* d_in, const int* in_sizes, int n_in,
                              void* d_out, int out_size, void* d_ws, size_t ws_size,
                              hipStream_t stream) {
  (void)in_sizes; (void)n_in; (void)out_size;
  const float* heat = (const float*)d_in[0];  // (16,1,480,640) f32
  const float* desc = (const float*)d_in[1];  // (16,256,60,80) f32
  float* out = (float*)d_out;

  // d_out layout (flat f32): kpts[16*400*2] | scores[16*400] | sampled[16*256*400] | heatmap[16*480*640]
  const size_t kpts_off = 0;
  const size_t sampled_off = (size_t)NB * KTOP * 2 + (size_t)NB * KTOP;  // 19200
  const size_t heat_off = sampled_off + (size_t)NB * CDESC * KTOP;       // 1657600

  // workspace: [0,256) = 16 counters (zeroed per call); then per-batch u64 candidate buffers
  unsigned* counters = (unsigned*)d_ws;
  unsigned long long* cand = (unsigned long long*)((char*)d_ws + 256);
  int cap = SORTN;  // power of two; expected candidates ~ HW/49 ~= 6.3K per batch
  while (cap > 1024 && (size_t)256 + (size_t)NB * (size_t)cap * 8 > ws_size) cap >>= 1;

  hipMemsetAsync(d_ws, 0, 256, stream);

  nms_kernel<<<dim3(300, NB), 256, 0, stream>>>(heat, counters, cand, cap);
  topk_sort_kernel<<<NB, 256, 0, stream>>>(heat, counters, cand, cap, out + kpts_off);
  sample_kernel<<<NB * KTOP, 256, 0, stream>>>(desc, out + kpts_off, out + sampled_off);
  hipMemcpyAsync(out + heat_off, heat, (size_t)NB * HWSZ * sizeof(float),
                 hipMemcpyDeviceToDevice, stream);
}